// CodeExtractorFromZ_79035988181261
// MI455X (gfx1250) — compile-verified
//
#include <hip/hip_runtime.h>

// Nearest-codebook extraction via V_WMMA_F32_16X16X4_F32 (CDNA5 / gfx1250).
//
//   quantized_z : (B=2, K*D=112, T=4096) f32   [d_in[0]]
//   codebooks   : (K=14, V=1024, D=8)    f32   [d_in[1]]
//   codes       : (B, K, T)              i32   [d_out]
//
// argmin_v (||c_v||^2 - 2 z.c_v)  ==  argmax_v (z.c_v - 0.5||c_v||^2).
// The -0.5||c||^2 bias is folded into the WMMA C accumulator, so the per-tile
// epilogue is only a running max + index select.

typedef __attribute__((ext_vector_type(2))) float v2f;
typedef __attribute__((ext_vector_type(8))) float v8f;

#define NCB    14
#define CBD    8
#define VOCAB  1024
#define BATCH  2
#define TIME   4096
#define TCHUNK 128          // t values per block = 8 waves * 16 rows

__global__ __launch_bounds__(256)
void codes_wmma_kernel(const float* __restrict__ z,
                       const float* __restrict__ cb,
                       int* __restrict__ out) {
  __shared__ float negHalfCsq[VOCAB];

  const int tid    = threadIdx.x;
  const int bk     = blockIdx.x / (TIME / TCHUNK);   // 0 .. B*K-1
  const int tchunk = blockIdx.x % (TIME / TCHUNK);
  const int k      = bk % NCB;

  const float* __restrict__ ck = cb + (size_t)k * VOCAB * CBD;

  // Precompute -0.5*||c_v||^2 into LDS (4 KB), shared by the 8 waves.
  for (int i = tid; i < VOCAB; i += 256) {
    const float* cv = ck + i * CBD;
    float s = 0.f;
#pragma unroll
    for (int d = 0; d < CBD; ++d) s += cv[d] * cv[d];
    negHalfCsq[i] = -0.5f * s;
  }
  __syncthreads();

  const int wave  = tid >> 5;
  const int lane  = tid & 31;
  const int half  = lane >> 4;        // K-split half of the wave
  const int l16   = lane & 15;
  const int tb    = tchunk * TCHUNK + wave * 16;
  const int dbase = half * 2;         // lanes 0-15: K=0,1 ; lanes 16-31: K=2,3

  // A fragment (16t x 8d), reused across the whole vocab scan.
  // z layout (B,K,D,T): element (d,t) at zb[d*TIME + t].
  const float* __restrict__ zb = z + (size_t)bk * CBD * TIME;
  const int t = tb + l16;
  v2f a0, a1;
  a0.x = zb[(dbase + 0) * TIME + t];
  a0.y = zb[(dbase + 1) * TIME + t];
  a1.x = zb[(dbase + 4) * TIME + t];
  a1.y = zb[(dbase + 5) * TIME + t];

  float bestv[8];
  int   besti[8];
#pragma unroll
  for (int r = 0; r < 8; ++r) { bestv[r] = -__builtin_inff(); besti[r] = 0; }

  // Scan 64 vocab tiles of 16; 2 chained 16x16x4 f32 WMMAs cover K=D=8.
  for (int vt = 0; vt < VOCAB / 16; ++vt) {
    const int v = vt * 16 + l16;                       // this lane's N (vocab id)
    const float* crow = ck + v * CBD + dbase;          // 8-byte aligned pair
    v2f b0 = *(const v2f*)(crow);                      // B[K=dbase..+1, N=v]
    v2f b1 = *(const v2f*)(crow + 4);                  // B[K=dbase+4..+5, N=v]

    const float cinit = negHalfCsq[v];                 // broadcast down column N=v
    v8f c;
#pragma unroll
    for (int r = 0; r < 8; ++r) c[r] = cinit;

    c = __builtin_amdgcn_wmma_f32_16x16x4_f32(false, a0, false, b0,
                                              (short)0, c, false, false);
    c = __builtin_amdgcn_wmma_f32_16x16x4_f32(false, a1, false, b1,
                                              (short)0, c, false, false);

    // Running argmax per (row r, lane-column). Branchless selects keep EXEC
    // all-ones around the WMMAs.
#pragma unroll
    for (int r = 0; r < 8; ++r) {
      const bool gt = c[r] > bestv[r];
      bestv[r] = gt ? c[r] : bestv[r];
      besti[r] = gt ? v    : besti[r];
    }
  }

  // Butterfly argmax across the 16 lanes of each half (masks 1,2,4,8 stay
  // inside a half under wave32). Tie-break: lowest vocab index wins, matching
  // argmin-first semantics.
#pragma unroll
  for (int r = 0; r < 8; ++r) {
#pragma unroll
    for (int off = 1; off < 16; off <<= 1) {
      const float ov = __shfl_xor(bestv[r], off, 32);
      const int   oi = __shfl_xor(besti[r], off, 32);
      const bool take = (ov > bestv[r]) || (ov == bestv[r] && oi < besti[r]);
      bestv[r] = take ? ov : bestv[r];
      besti[r] = take ? oi : besti[r];
    }
  }

  // C/D layout: VGPR r holds row M=r (lanes 0-15) and M=r+8 (lanes 16-31).
  int* __restrict__ ob = out + (size_t)bk * TIME + tb;
  if (l16 == 0) {
#pragma unroll
    for (int r = 0; r < 8; ++r) ob[half * 8 + r] = besti[r];
  }
}

extern "C" void kernel_launch(void* const* d_in, const int* in_sizes, int n_in,
                              void* d_out, int out_size, void* d_ws, size_t ws_size,
                              hipStream_t stream) {
  (void)in_sizes; (void)n_in; (void)out_size; (void)d_ws; (void)ws_size;
  const float* z  = (const float*)d_in[0];
  const float* cb = (const float*)d_in[1];
  int* out        = (int*)d_out;

  const int grid = BATCH * NCB * (TIME / TCHUNK);   // 896 blocks
  codes_wmma_kernel<<<grid, 256, 0, stream>>>(z, cb, out);
}